// AttModel_38525856645267
// MI455X (gfx1250) — compile-verified
//
#include <hip/hip_runtime.h>

// ---------------------------------------------------------------------------
// MI455X (gfx1250) fused attention, fp32 end-to-end via V_WMMA_F32_16X16X4_F32.
//  B=8 N=1024 IN=512 H=8 DK=DV=64 OUT=512
//  d_out = [ out (8*1024*512) | att_weights_raw (8*8*1024*1024) ]  (fp32)
//  d_ws  = q | k | v (each B,H,N,64) | attn_out (B,N,512)  -> 64 MB fp32
// ---------------------------------------------------------------------------

typedef float v2f __attribute__((ext_vector_type(2)));
typedef float v8f __attribute__((ext_vector_type(8)));

#define WMMA_F32(a, b, c) \
  __builtin_amdgcn_wmma_f32_16x16x4_f32(false, (a), false, (b), (short)0, (c), false, false)

namespace {
constexpr int Bb = 8, Nn = 1024, INc = 512, Hh = 8, Dd = 64, OUTc = 512;
constexpr float SCALE = 0.125f;          // 1/sqrt(64)
constexpr int SROW = 1026;               // LDS row stride (floats): 8B aligned, conflict-free
constexpr int ATTN_LDS_FLOATS = 32 * SROW + 32 * 8 + 32;
}

// ===========================================================================
// GEMM: C[128 x 64 tile] = relu(A[8192x512] @ W[512x512] + bias)
// block = 256 thr (8 waves); wave w -> rows [bm*128+w*16, +16), cols [bn*64,+64)
// scatter store into (B,H,N,64) layout (head-major) for the attention kernel.
// ===========================================================================
__global__ __launch_bounds__(256) void proj_qkv_kernel(
    const float* __restrict__ x, const float* __restrict__ W,
    const float* __restrict__ bias, float* __restrict__ outBHND) {
  const int lane = threadIdx.x & 31;
  const int wave = threadIdx.x >> 5;
  const int l15 = lane & 15;
  const int hl = lane >> 4;       // which 16-lane half
  const int hl2 = hl << 1;
  const int row0 = blockIdx.x * 128 + wave * 16;
  const int col0 = blockIdx.y * 64;

  v8f acc0 = {}, acc1 = {}, acc2 = {}, acc3 = {};
  const float* arow = x + (size_t)(row0 + l15) * INc;
  for (int kk = 0; kk < INc / 4; ++kk) {
    v2f a = *(const v2f*)(arow + kk * 4 + hl2);                 // A frag (16x4)
    const float* wp = W + (size_t)(kk * 4 + hl2) * OUTc + col0 + l15;
    v2f b0, b1, b2, b3;                                         // B frags (4x16)
    b0[0] = wp[0];  b0[1] = wp[OUTc];
    b1[0] = wp[16]; b1[1] = wp[OUTc + 16];
    b2[0] = wp[32]; b2[1] = wp[OUTc + 32];
    b3[0] = wp[48]; b3[1] = wp[OUTc + 48];
    acc0 = WMMA_F32(a, b0, acc0);
    acc1 = WMMA_F32(a, b1, acc1);
    acc2 = WMMA_F32(a, b2, acc2);
    acc3 = WMMA_F32(a, b3, acc3);
  }

  v8f accs[4] = {acc0, acc1, acc2, acc3};
  const int mbase = row0 + (hl << 3);
#pragma unroll
  for (int nt = 0; nt < 4; ++nt) {
    const int col = col0 + nt * 16 + l15;
    const float bia = bias[col];
    const int h = col >> 6, d = col & 63;
#pragma unroll
    for (int vv = 0; vv < 8; ++vv) {
      const int gm = mbase + vv;                 // global row in 0..8191
      float val = accs[nt][vv] + bia;
      val = val > 0.f ? val : 0.f;               // relu
      const int b = gm >> 10, nrow = gm & 1023;
      outBHND[((size_t)(b * Hh + h) * Nn + nrow) * Dd + d] = val;
    }
  }
}

// ===========================================================================
// Fused attention per (b, h, 32-query tile):
//  S = q@k^T*scale  -> raw to d_out;  mask -> LDS;  softmax in LDS;
//  O = P@V * 1/L + v_residual -> attn_out (B,N,512)
// ===========================================================================
__global__ __launch_bounds__(256) void attn_kernel(
    const float* __restrict__ q, const float* __restrict__ k,
    const float* __restrict__ v, const int* __restrict__ mask,
    float* __restrict__ attw, float* __restrict__ attn_out) {
  extern __shared__ float smem[];
  float* S = smem;                        // 32 x SROW masked scores / probs
  float* redbuf = smem + 32 * SROW;       // 32 x 8 reduction scratch
  float* invl = redbuf + 32 * 8;          // 32 reciprocal row sums

  const int tid = threadIdx.x;
  const int lane = tid & 31, wave = tid >> 5;
  const int l15 = lane & 15, hl = lane >> 4, hl2 = hl << 1;
  const int qt = blockIdx.x & 31;
  const int h = (blockIdx.x >> 5) & 7;
  const int b = blockIdx.x >> 8;
  const int q0 = qt * 32;
  const size_t head = (size_t)(b * Hh + h) * Nn;   // row base in (B*H*N)
  const float* qb = q + head * Dd;
  const float* kb = k + head * Dd;
  const float* vb = v + head * Dd;

  // -------- Phase 1: scores. wave -> rows [(w&1)*16,+16), keys [(w>>1)*256,+256)
  {
    const int r0 = (wave & 1) * 16;
    const int cg = wave >> 1;
    v2f qa[16];                                        // Q as 16 A-frags (K=64)
    const float* qrow = qb + (size_t)(q0 + r0 + l15) * Dd + hl2;
#pragma unroll
    for (int kk = 0; kk < 16; ++kk) qa[kk] = *(const v2f*)(qrow + kk * 4);

    const int mrow_base = r0 + (hl << 3);
    for (int j = 0; j < 16; ++j) {
      const int key = cg * 256 + j * 16 + l15;
      const float* kp = kb + (size_t)key * Dd + hl2;   // B frag: k row (L2-hot)
      v8f acc = {};
#pragma unroll
      for (int kk = 0; kk < 16; ++kk) {
        v2f bf = *(const v2f*)(kp + kk * 4);
        acc = WMMA_F32(qa[kk], bf, acc);
      }
#pragma unroll
      for (int vv = 0; vv < 8; ++vv) {
        const int m = mrow_base + vv;                  // local row 0..31
        const float s = acc[vv] * SCALE;
        attw[(head + (q0 + m)) * (size_t)Nn + key] = s;          // raw output
        const int mk = mask[((size_t)b * Nn + (q0 + m)) * Nn + key];
        S[m * SROW + key] = mk ? s : -1e9f;
      }
    }
  }
  __syncthreads();

  // -------- Phase 2: row softmax in LDS (8 threads x 128 cols per row)
  {
    const int row = tid & 31, seg = tid >> 5;
    float* srow = S + row * SROW + seg * 128;
    float mx = -3.4e38f;
    for (int i = 0; i < 128; ++i) mx = fmaxf(mx, srow[i]);
    redbuf[row * 8 + seg] = mx;
    __syncthreads();
    float rmax = redbuf[row * 8];
#pragma unroll
    for (int i = 1; i < 8; ++i) rmax = fmaxf(rmax, redbuf[row * 8 + i]);
    __syncthreads();
    float sum = 0.f;
    for (int i = 0; i < 128; ++i) {
      const float e = __expf(srow[i] - rmax);
      srow[i] = e;
      sum += e;
    }
    redbuf[row * 8 + seg] = sum;
    __syncthreads();
    if (seg == 0) {
      float t = 0.f;
#pragma unroll
      for (int i = 0; i < 8; ++i) t += redbuf[row * 8 + i];
      invl[row] = 1.f / t;                 // all-masked row -> 1024 exp(0) = uniform, matches ref
    }
  }
  __syncthreads();

  // -------- Phase 3: O = (P * invl) @ V + v residual. wave -> 16 rows x 16 dv cols
  {
    const int r0 = (wave & 1) * 16;
    const int c0 = (wave >> 1) * 16;
    v8f acc = {};
    const float* arow = S + (size_t)(r0 + l15) * SROW + hl2;
    for (int kk = 0; kk < 256; ++kk) {     // K = 1024 keys, step 4
      v2f a = *(const v2f*)(arow + kk * 4);
      const int kr = kk * 4 + hl2;
      v2f bf;
      bf[0] = vb[(size_t)kr * Dd + c0 + l15];
      bf[1] = vb[(size_t)(kr + 1) * Dd + c0 + l15];
      acc = WMMA_F32(a, bf, acc);
    }
    const int mb = r0 + (hl << 3);
    const int dv = c0 + l15;
#pragma unroll
    for (int vv = 0; vv < 8; ++vv) {
      const int m = mb + vv;
      const float o = acc[vv] * invl[m] + vb[(size_t)(q0 + m) * Dd + dv];
      attn_out[((size_t)(b * Nn) + (q0 + m)) * (Hh * Dd) + h * Dd + dv] = o;
    }
  }
}

// ===========================================================================
// out = relu(attn_out[8192x512] @ Wo[512x512] + bo) -> d_out part 1 (row-major)
// ===========================================================================
__global__ __launch_bounds__(256) void proj_out_kernel(
    const float* __restrict__ A, const float* __restrict__ Wo,
    const float* __restrict__ bo, float* __restrict__ out) {
  const int lane = threadIdx.x & 31;
  const int wave = threadIdx.x >> 5;
  const int l15 = lane & 15;
  const int hl = lane >> 4;
  const int hl2 = hl << 1;
  const int row0 = blockIdx.x * 128 + wave * 16;
  const int col0 = blockIdx.y * 64;

  v8f acc0 = {}, acc1 = {}, acc2 = {}, acc3 = {};
  const float* arow = A + (size_t)(row0 + l15) * (Hh * Dd);
  for (int kk = 0; kk < (Hh * Dd) / 4; ++kk) {
    v2f a = *(const v2f*)(arow + kk * 4 + hl2);
    const float* wp = Wo + (size_t)(kk * 4 + hl2) * OUTc + col0 + l15;
    v2f b0, b1, b2, b3;
    b0[0] = wp[0];  b0[1] = wp[OUTc];
    b1[0] = wp[16]; b1[1] = wp[OUTc + 16];
    b2[0] = wp[32]; b2[1] = wp[OUTc + 32];
    b3[0] = wp[48]; b3[1] = wp[OUTc + 48];
    acc0 = WMMA_F32(a, b0, acc0);
    acc1 = WMMA_F32(a, b1, acc1);
    acc2 = WMMA_F32(a, b2, acc2);
    acc3 = WMMA_F32(a, b3, acc3);
  }

  v8f accs[4] = {acc0, acc1, acc2, acc3};
  const int mbase = row0 + (hl << 3);
#pragma unroll
  for (int nt = 0; nt < 4; ++nt) {
    const int col = col0 + nt * 16 + l15;
    const float bia = bo[col];
#pragma unroll
    for (int vv = 0; vv < 8; ++vv) {
      const int gm = mbase + vv;
      float val = accs[nt][vv] + bia;
      val = val > 0.f ? val : 0.f;
      out[(size_t)gm * OUTc + col] = val;
    }
  }
}

// ===========================================================================
extern "C" void kernel_launch(void* const* d_in, const int* in_sizes, int n_in,
                              void* d_out, int out_size, void* d_ws, size_t ws_size,
                              hipStream_t stream) {
  (void)in_sizes; (void)n_in; (void)out_size; (void)ws_size;
  const float* x  = (const float*)d_in[0];
  const float* Wv = (const float*)d_in[1];
  const float* bv = (const float*)d_in[2];
  const float* Wk = (const float*)d_in[3];
  const float* bk = (const float*)d_in[4];
  const float* Wq = (const float*)d_in[5];
  const float* bq = (const float*)d_in[6];
  const float* Wo = (const float*)d_in[7];
  const float* bo = (const float*)d_in[8];
  const int* mask = (const int*)d_in[9];

  float* out  = (float*)d_out;
  float* attw = out + (size_t)Bb * Nn * OUTc;          // raw attention weights

  const size_t qkv_elems = (size_t)Bb * Hh * Nn * Dd;  // 4,194,304 each
  float* qws = (float*)d_ws;
  float* kws = qws + qkv_elems;
  float* vws = kws + qkv_elems;
  float* aows = vws + qkv_elems;                       // (B,N,512) attention output

  const dim3 gproj(64, 8);                             // 8192/128 x 512/64
  proj_qkv_kernel<<<gproj, 256, 0, stream>>>(x, Wq, bq, qws);
  proj_qkv_kernel<<<gproj, 256, 0, stream>>>(x, Wk, bk, kws);
  proj_qkv_kernel<<<gproj, 256, 0, stream>>>(x, Wv, bv, vws);

  attn_kernel<<<Bb * Hh * (Nn / 32), 256, ATTN_LDS_FLOATS * sizeof(float), stream>>>(
      qws, kws, vws, mask, attw, aows);                // ~132 KB LDS, fine on 320 KB WGP

  proj_out_kernel<<<gproj, 256, 0, stream>>>(aows, Wo, bo, out);
}